// MambaLMHeadModel_3642132267308
// MI455X (gfx1250) — compile-verified
//
#include <hip/hip_runtime.h>
#include <hip/hip_bf16.h>

#define D_IN   768
#define D_M    1536
#define NSTATE 16
#define DTRANK 48
#define KCONV  4
#define NLAYER 12
#define VOCABP 50280
#define SEQ    512
#define XDBL_W (DTRANK + 2 * NSTATE)   // 80

#define KC        64                   // k-chunk per TDM transfer (floats)
#define SW_STRIDE 68                   // LDS row stride: 64 + 4 pad DWORDs (bank-conflict-free)
#define SW_FLOATS (64 * SW_STRIDE)     // one LDS buffer (17408 B)

typedef __attribute__((ext_vector_type(2))) float v2f;
typedef __attribute__((ext_vector_type(8))) float v8f;
typedef unsigned int u32x4 __attribute__((ext_vector_type(4)));
typedef int          i32x4 __attribute__((ext_vector_type(4)));
typedef int          i32x8 __attribute__((ext_vector_type(8)));

__device__ __forceinline__ float sigmoidf_(float x) { return 1.f / (1.f + __expf(-x)); }
__device__ __forceinline__ float siluf_(float x)    { return x * sigmoidf_(x); }
__device__ __forceinline__ float softplusf_(float x){ return (x > 20.f) ? x : log1pf(__expf(x)); }

// ---------------------------------------------------------------------------
// TDM: issue one 2-D tile load (rows x KC floats, row stride ldw) into LDS.
// D# per CDNA5 ISA ch.8: group0 = {count, lds_addr, global_addr, type=2};
// group1 = {data_size=4B, pad 4 DWORDs every 64 DWORDs, dims/tile/stride}.
// ---------------------------------------------------------------------------
__device__ __forceinline__ void tdm_issue(unsigned lds_byte, const float* g,
                                          int rows, int ldw)
{
    unsigned long long ga = (unsigned long long)(size_t)g;
    u32x4 g0;
    g0.x = 1u;                                             // count=1, user descriptor
    g0.y = lds_byte;                                       // lds_addr
    g0.z = (unsigned)(ga & 0xffffffffu);                   // global_addr[31:0]
    g0.w = (unsigned)((ga >> 32) & 0x01ffffffu) | (2u << 30); // addr[56:32] | type=2
    i32x8 g1;
    g1[0] = (int)((2u << 16)                               // data_size = 4 bytes
                | (1u << 20)                               // pad_enable
                | (5u << 22)                               // pad_interval: every 64 DWORDs
                | (3u << 25));                             // pad_amount: 4 DWORDs
    g1[1] = (int)((unsigned)KC << 16);                     // tensor_dim0[15:0] = KC
    g1[2] = (int)((unsigned)rows << 16);                   // tensor_dim1[15:0] = rows
    g1[3] = (int)((unsigned)KC << 16);                     // tile_dim0 = KC
    g1[4] = (int)(rows & 0xffff);                          // tile_dim1 = rows
    g1[5] = ldw;                                           // tensor_dim0_stride[31:0]
    g1[6] = 0;
    g1[7] = 0;
    i32x4 z4 = {0, 0, 0, 0};                               // 2-D tensor: groups 2/3 unused
#if __clang_major__ >= 23
    i32x8 z8 = {0, 0, 0, 0, 0, 0, 0, 0};
    __builtin_amdgcn_tensor_load_to_lds(g0, g1, z4, z4, z8, 0);
#else
    __builtin_amdgcn_tensor_load_to_lds(g0, g1, z4, z4, 0);
#endif
}

// ---------------------------------------------------------------------------
// TDM-staged fp32 WMMA GEMM:  C[M,N] = A[M,K] * W[N,K]^T  (row-major).
// Block = 8 waves = 128(M) x 64(N) tile. The shared 64-row W tile is streamed
// into LDS by the Tensor Data Mover (double-buffered, TENSORcnt + barrier),
// so the 8 waves read it once from LDS instead of 8x from cache.
// Requires K % KC == 0 and M % 128 == 0.
// EPI: 0 = store, 2 = C += acc
// ---------------------------------------------------------------------------
template <int EPI>
__global__ __launch_bounds__(256) void gemm_wmma_f32_tdm(
    const float* __restrict__ A, const float* __restrict__ W,
    float* __restrict__ C, int M, int N, int K, int lda, int ldw, int ldc)
{
    __shared__ float sw[2][SW_FLOATS];                     // only LDS object -> offset 0

    const int lane = threadIdx.x & 31;
    const int wib  = threadIdx.x >> 5;                     // wave in block (0..7)
    const int mblocks = M >> 7;
    const int mb = blockIdx.x % mblocks;
    const int nt = blockIdx.x / mblocks;
    const int n0 = nt << 6;
    const int m0 = (mb << 7) + (wib << 4);
    const int r  = lane & 15;
    const int hi = lane >> 4;
    const int kb = hi << 1;

    const int rows = min(64, N - n0);                      // valid W rows in this column tile
    const float* __restrict__ arow  = A + (size_t)(m0 + r) * lda + kb;
    const float* __restrict__ wtile = W + (size_t)n0 * ldw;

    if (threadIdx.x < 32)                                  // wave 0 drives the TDM
        tdm_issue(0u, wtile, rows, ldw);
    __builtin_amdgcn_s_wait_tensorcnt(0);
    __syncthreads();

    v8f acc0 = {}, acc1 = {}, acc2 = {}, acc3 = {};
    int cur = 0;
    for (int kc = 0; kc < K; kc += KC) {
        if (kc + KC < K && threadIdx.x < 32)               // prefetch next chunk via TDM
            tdm_issue((unsigned)((cur ^ 1) * (SW_FLOATS * 4)), wtile + kc + KC, rows, ldw);

        const float* __restrict__ aP = arow + kc;
        const float* __restrict__ sP = &sw[cur][kb + (size_t)r * SW_STRIDE];
        #pragma unroll
        for (int kk = 0; kk < KC / 4; kk++) {
            v2f a;  a.x  = aP[kk * 4];                 a.y  = aP[kk * 4 + 1];
            const float* s0 = sP + kk * 4;
            const float* s1 = s0 + 16 * SW_STRIDE;
            const float* s2 = s0 + 32 * SW_STRIDE;
            const float* s3 = s0 + 48 * SW_STRIDE;
            v2f b0; b0.x = s0[0]; b0.y = s0[1];
            v2f b1; b1.x = s1[0]; b1.y = s1[1];
            v2f b2; b2.x = s2[0]; b2.y = s2[1];
            v2f b3; b3.x = s3[0]; b3.y = s3[1];
            acc0 = __builtin_amdgcn_wmma_f32_16x16x4_f32(false, a, false, b0, (short)0, acc0, false, false);
            acc1 = __builtin_amdgcn_wmma_f32_16x16x4_f32(false, a, false, b1, (short)0, acc1, false, false);
            acc2 = __builtin_amdgcn_wmma_f32_16x16x4_f32(false, a, false, b2, (short)0, acc2, false, false);
            acc3 = __builtin_amdgcn_wmma_f32_16x16x4_f32(false, a, false, b3, (short)0, acc3, false, false);
        }
        __builtin_amdgcn_s_wait_tensorcnt(0);              // next chunk landed (wave 0; others no-op)
        __syncthreads();                                   // publish buffer + protect overwrite
        cur ^= 1;
    }

    #pragma unroll
    for (int v = 0; v < 8; v++) {
        const int m = m0 + v + (hi << 3);
        float* __restrict__ crow = C + (size_t)m * ldc;
        float vals[4] = { acc0[v], acc1[v], acc2[v], acc3[v] };
        #pragma unroll
        for (int j = 0; j < 4; j++) {
            const int c = n0 + j * 16 + r;
            if (c < N) {
                if (EPI == 0) crow[c]  = vals[j];
                else          crow[c] += vals[j];
            }
        }
    }
}

// ---------------------------------------------------------------------------
// Direct-load fp32 WMMA GEMM (used where K % 64 != 0, i.e. the delta proj).
// EPI: 1 = store softplus(acc + bias[col])
// ---------------------------------------------------------------------------
template <int EPI>
__global__ __launch_bounds__(256) void gemm_wmma_f32(
    const float* __restrict__ A, const float* __restrict__ W,
    float* __restrict__ C, const float* __restrict__ bias,
    int M, int N, int K, int lda, int ldw, int ldc)
{
    const int lane = threadIdx.x & 31;
    const int wid  = blockIdx.x * (blockDim.x >> 5) + (threadIdx.x >> 5);
    const int mtiles   = M >> 4;
    const int ntiles64 = (N + 63) >> 6;
    if (wid >= mtiles * ntiles64) return;

    const int mt = wid % mtiles;
    const int nt = wid / mtiles;
    const int m0 = mt << 4;
    const int n0 = nt << 6;
    const int r  = lane & 15;
    const int hi = lane >> 4;
    const int kb = hi << 1;

    const float* __restrict__ arow = A + (size_t)(m0 + r) * lda + kb;
    const int c0 = n0 + r, c1 = n0 + 16 + r, c2 = n0 + 32 + r, c3 = n0 + 48 + r;
    const float* __restrict__ w0 = W + (size_t)(c0 < N ? c0 : 0) * ldw + kb;
    const float* __restrict__ w1 = W + (size_t)(c1 < N ? c1 : 0) * ldw + kb;
    const float* __restrict__ w2 = W + (size_t)(c2 < N ? c2 : 0) * ldw + kb;
    const float* __restrict__ w3 = W + (size_t)(c3 < N ? c3 : 0) * ldw + kb;

    v8f acc0 = {}, acc1 = {}, acc2 = {}, acc3 = {};
    v2f a_c, b0_c, b1_c, b2_c, b3_c;
    a_c.x  = arow[0]; a_c.y  = arow[1];
    b0_c.x = w0[0];   b0_c.y = w0[1];
    b1_c.x = w1[0];   b1_c.y = w1[1];
    b2_c.x = w2[0];   b2_c.y = w2[1];
    b3_c.x = w3[0];   b3_c.y = w3[1];

    for (int k = 4; k < K; k += 4) {
        v2f a_n, b0_n, b1_n, b2_n, b3_n;
        a_n.x  = arow[k]; a_n.y  = arow[k + 1];
        b0_n.x = w0[k];   b0_n.y = w0[k + 1];
        b1_n.x = w1[k];   b1_n.y = w1[k + 1];
        b2_n.x = w2[k];   b2_n.y = w2[k + 1];
        b3_n.x = w3[k];   b3_n.y = w3[k + 1];

        acc0 = __builtin_amdgcn_wmma_f32_16x16x4_f32(false, a_c, false, b0_c, (short)0, acc0, false, false);
        acc1 = __builtin_amdgcn_wmma_f32_16x16x4_f32(false, a_c, false, b1_c, (short)0, acc1, false, false);
        acc2 = __builtin_amdgcn_wmma_f32_16x16x4_f32(false, a_c, false, b2_c, (short)0, acc2, false, false);
        acc3 = __builtin_amdgcn_wmma_f32_16x16x4_f32(false, a_c, false, b3_c, (short)0, acc3, false, false);
        a_c = a_n; b0_c = b0_n; b1_c = b1_n; b2_c = b2_n; b3_c = b3_n;
    }
    acc0 = __builtin_amdgcn_wmma_f32_16x16x4_f32(false, a_c, false, b0_c, (short)0, acc0, false, false);
    acc1 = __builtin_amdgcn_wmma_f32_16x16x4_f32(false, a_c, false, b1_c, (short)0, acc1, false, false);
    acc2 = __builtin_amdgcn_wmma_f32_16x16x4_f32(false, a_c, false, b2_c, (short)0, acc2, false, false);
    acc3 = __builtin_amdgcn_wmma_f32_16x16x4_f32(false, a_c, false, b3_c, (short)0, acc3, false, false);

    #pragma unroll
    for (int v = 0; v < 8; v++) {
        const int m = m0 + v + (hi << 3);
        float* __restrict__ crow = C + (size_t)m * ldc;
        float vals[4] = { acc0[v], acc1[v], acc2[v], acc3[v] };
        int   cols[4] = { c0, c1, c2, c3 };
        #pragma unroll
        for (int j = 0; j < 4; j++) {
            if (cols[j] < N) {
                if (EPI == 0)      crow[cols[j]]  = vals[j];
                else if (EPI == 1) crow[cols[j]]  = softplusf_(vals[j] + bias[cols[j]]);
                else               crow[cols[j]] += vals[j];
            }
        }
    }
}

// ---------------------------------------------------------------------------
// Elementwise / recurrence kernels
// ---------------------------------------------------------------------------
__global__ void embed_k(const int* __restrict__ ids, const float* __restrict__ emb,
                        float* __restrict__ x)
{
    int i = blockIdx.x * blockDim.x + threadIdx.x;
    if (i >= SEQ * D_IN) return;
    int t = i / D_IN, c = i - t * D_IN;
    int id = ids[t];
    if (id < 0) id = 0; if (id >= VOCABP) id = VOCABP - 1;
    x[i] = emb[(size_t)id * D_IN + c];
}

__global__ __launch_bounds__(256) void rmsnorm_k(const float* __restrict__ x,
                                                 const float* __restrict__ w,
                                                 float* __restrict__ y, int D)
{
    __shared__ float red[8];
    const int row = blockIdx.x;
    const float* __restrict__ xr = x + (size_t)row * D;
    float s = 0.f;
    for (int c = threadIdx.x; c < D; c += blockDim.x) { float v = xr[c]; s += v * v; }
    #pragma unroll
    for (int off = 16; off; off >>= 1) s += __shfl_down(s, off, 32);
    if ((threadIdx.x & 31) == 0) red[threadIdx.x >> 5] = s;
    __syncthreads();
    if (threadIdx.x < 8) {
        s = red[threadIdx.x];
        #pragma unroll
        for (int off = 4; off; off >>= 1) s += __shfl_down(s, off, 8);
        if (threadIdx.x == 0) red[0] = s;
    }
    __syncthreads();
    const float inv = rsqrtf(red[0] / (float)D + 1e-5f);
    for (int c = threadIdx.x; c < D; c += blockDim.x)
        y[(size_t)row * D + c] = xr[c] * inv * w[c];
}

__global__ void conv_silu_k(const float* __restrict__ xz, const float* __restrict__ cw,
                            const float* __restrict__ cb, float* __restrict__ xc)
{
    int i = blockIdx.x * blockDim.x + threadIdx.x;
    if (i >= SEQ * D_M) return;
    int t = i / D_M, d = i - t * D_M;
    float acc = cb[d];
    #pragma unroll
    for (int k = 0; k < KCONV; k++) {
        int tt = t - (KCONV - 1) + k;
        if (tt >= 0) acc += xz[(size_t)tt * (2 * D_M) + d] * cw[d * KCONV + k];
    }
    xc[i] = siluf_(acc);
}

__global__ __launch_bounds__(256) void scan_k(
    const float* __restrict__ xz,   const float* __restrict__ xc,
    const float* __restrict__ xdbl, const float* __restrict__ delta,
    const float* __restrict__ A_log,const float* __restrict__ Dsk,
    float* __restrict__ yg)
{
    int d = blockIdx.x * blockDim.x + threadIdx.x;
    if (d >= D_M) return;
    float Ad[NSTATE], h[NSTATE];
    #pragma unroll
    for (int n = 0; n < NSTATE; n++) { Ad[n] = -__expf(A_log[d * NSTATE + n]); h[n] = 0.f; }
    const float dskip = Dsk[d];
    for (int t = 0; t < SEQ; t++) {
        const float dlt = delta[(size_t)t * D_M + d];
        const float xcv = xc[(size_t)t * D_M + d];
        const float du  = dlt * xcv;
        const float* __restrict__ Bt = xdbl + (size_t)t * XDBL_W + DTRANK;
        const float* __restrict__ Ct = Bt + NSTATE;
        float y = 0.f;
        #pragma unroll
        for (int n = 0; n < NSTATE; n++) {
            h[n] = __expf(dlt * Ad[n]) * h[n] + du * Bt[n];
            y = fmaf(h[n], Ct[n], y);
        }
        const float zv = xz[(size_t)t * (2 * D_M) + D_M + d];
        yg[(size_t)t * D_M + d] = (y + xcv * dskip) * siluf_(zv);
    }
}

// ---------------------------------------------------------------------------
template <int EPI>
static void launch_gemm_tdm(const float* A, const float* W, float* C,
                            int M, int N, int K, int lda, int ldw, int ldc, hipStream_t s)
{
    int blocks = (M >> 7) * ((N + 63) >> 6);
    gemm_wmma_f32_tdm<EPI><<<blocks, 256, 0, s>>>(A, W, C, M, N, K, lda, ldw, ldc);
}

template <int EPI>
static void launch_gemm(const float* A, const float* W, float* C, const float* bias,
                        int M, int N, int K, int lda, int ldw, int ldc, hipStream_t s)
{
    int waves  = (M >> 4) * ((N + 63) >> 6);
    int blocks = (waves + 7) / 8;
    gemm_wmma_f32<EPI><<<blocks, 256, 0, s>>>(A, W, C, bias, M, N, K, lda, ldw, ldc);
}

extern "C" void kernel_launch(void* const* d_in, const int* in_sizes, int n_in,
                              void* d_out, int out_size, void* d_ws, size_t ws_size,
                              hipStream_t stream)
{
    const int*   ids      = (const int*)  d_in[0];
    const float* emb      = (const float*)d_in[1];
    const float* W_in     = (const float*)d_in[2];
    const float* conv_w   = (const float*)d_in[3];
    const float* conv_b   = (const float*)d_in[4];
    const float* W_x      = (const float*)d_in[5];
    const float* W_dt     = (const float*)d_in[6];
    const float* b_dt     = (const float*)d_in[7];
    const float* A_log    = (const float*)d_in[8];
    const float* D_skip   = (const float*)d_in[9];
    const float* W_out    = (const float*)d_in[10];
    const float* norm_w   = (const float*)d_in[11];
    const float* norm_f_w = (const float*)d_in[12];
    float* out = (float*)d_out;

    float* ws   = (float*)d_ws;
    size_t o    = 0;
    float* x    = ws + o;  o += (size_t)SEQ * D_IN;
    float* xn   = ws + o;  o += (size_t)SEQ * D_IN;
    float* xzb  = ws + o;  o += (size_t)SEQ * 2 * D_M;
    float* xc   = ws + o;  o += (size_t)SEQ * D_M;
    float* xdbl = ws + o;  o += (size_t)SEQ * XDBL_W;
    float* dlt  = ws + o;  o += (size_t)SEQ * D_M;
    float* yg   = ws + o;  o += (size_t)SEQ * D_M;

    embed_k<<<(SEQ * D_IN + 255) / 256, 256, 0, stream>>>(ids, emb, x);

    for (int l = 0; l < NLAYER; l++) {
        rmsnorm_k<<<SEQ, 256, 0, stream>>>(x, norm_w + (size_t)l * D_IN, xn, D_IN);

        // xz = xn @ W_in^T : (512 x 3072), K=768  (TDM-staged)
        launch_gemm_tdm<0>(xn, W_in + (size_t)l * 2 * D_M * D_IN, xzb,
                           SEQ, 2 * D_M, D_IN, D_IN, D_IN, 2 * D_M, stream);

        conv_silu_k<<<(SEQ * D_M + 255) / 256, 256, 0, stream>>>(
            xzb, conv_w + (size_t)l * D_M * KCONV, conv_b + (size_t)l * D_M, xc);

        // x_dbl = xc @ W_x^T : (512 x 80), K=1536  (TDM-staged)
        launch_gemm_tdm<0>(xc, W_x + (size_t)l * XDBL_W * D_M, xdbl,
                           SEQ, XDBL_W, D_M, D_M, D_M, XDBL_W, stream);

        // delta = softplus(x_dbl[:, :48] @ W_dt^T + b_dt) : K=48 (direct path)
        launch_gemm<1>(xdbl, W_dt + (size_t)l * D_M * DTRANK, dlt,
                       b_dt + (size_t)l * D_M,
                       SEQ, D_M, DTRANK, XDBL_W, DTRANK, D_M, stream);

        scan_k<<<(D_M + 255) / 256, 256, 0, stream>>>(
            xzb, xc, xdbl, dlt, A_log + (size_t)l * D_M * NSTATE,
            D_skip + (size_t)l * D_M, yg);

        // x += yg @ W_out^T : (512 x 768), K=1536  (TDM-staged, accumulate)
        launch_gemm_tdm<2>(yg, W_out + (size_t)l * D_IN * D_M, x,
                           SEQ, D_IN, D_M, D_M, D_M, D_IN, stream);
    }

    rmsnorm_k<<<SEQ, 256, 0, stream>>>(x, norm_f_w, xn, D_IN);

    // logits = xn @ embedding^T : (512 x 50280), K=768  (TDM-staged)
    launch_gemm_tdm<0>(xn, emb, out, SEQ, VOCABP, D_IN, D_IN, D_IN, VOCABP, stream);
}